// PatientPoolEncoder_16612933501060
// MI455X (gfx1250) — compile-verified
//
#include <hip/hip_runtime.h>
#include <hip/hip_bf16.h>
#include <math.h>

// Problem constants (match the reference).
#define N_ 32
#define L_ 2048
#define C_ 256
#define DEM_ 8
#define H1_ 40
#define H2_ 20
#define OC_ 788            // 3*C + 20
#define NCH_ 16            // chunks along L
#define CL_ 128            // chunk length (NCH_*CL_ == L_)

typedef __attribute__((ext_vector_type(2))) float v2f;
typedef __attribute__((ext_vector_type(8))) float v8f;

// ---------------------------------------------------------------------------
// Kernel 1: dem MLP with f32 WMMA (exact precision path).
//   d = relu(relu(dem @ w1 + b1) @ w2 + b2)     (32x8)->(32x40)->(32x20)
// One wave (32 threads). Uses V_WMMA_F32_16X16X4_F32 per ISA 7.12.2 layouts:
//   A 16x4 f32: lane l holds row M=l%16; VGPR0/1 = K {0,1} (lanes<16) or {2,3}
//   B 4x16 f32: lane l holds col N=l%16; VGPR0/1 = K {0,1} (lanes<16) or {2,3}
//   C/D 16x16:  VGPR v, lane l -> M = v + (l<16?0:8), N = l%16
// ---------------------------------------------------------------------------
__global__ __launch_bounds__(32) void mlp_wmma_kernel(
    const float* __restrict__ dem, const float* __restrict__ w1,
    const float* __restrict__ b1, const float* __restrict__ w2,
    const float* __restrict__ b2, float* __restrict__ dout) {
  __shared__ float h[N_][H1_];   // hidden activations

  const int l    = threadIdx.x;        // 0..31
  const int half = (l < 16) ? 0 : 1;   // which half-wave
  const int mrow = l & 15;
  const int ncol = l & 15;

  // ---- layer 1: (32x8) @ (8x40), tiles: M 2, N 3 (pad 40->48), K 2 ----
  for (int mt = 0; mt < 2; ++mt) {
    for (int nt = 0; nt < 3; ++nt) {
      v8f acc = {};
      const int col = nt * 16 + ncol;
      for (int k = 0; k < 2; ++k) {
        const int row = mt * 16 + mrow;
        const int kr  = k * 4 + half * 2;
        v2f a, b;
        a.x = dem[row * DEM_ + kr + 0];
        a.y = dem[row * DEM_ + kr + 1];
        b.x = (col < H1_) ? w1[(kr + 0) * H1_ + col] : 0.0f;
        b.y = (col < H1_) ? w1[(kr + 1) * H1_ + col] : 0.0f;
        acc = __builtin_amdgcn_wmma_f32_16x16x4_f32(
            false, a, false, b, (short)0, acc, false, false);
      }
      if (col < H1_) {
        const float bias = b1[col];
        for (int v = 0; v < 8; ++v) {
          const int row = mt * 16 + v + half * 8;
          h[row][col] = fmaxf(acc[v] + bias, 0.0f);
        }
      }
    }
  }
  __syncthreads();

  // ---- layer 2: (32x40) @ (40x20), tiles: M 2, N 2 (pad 20->32), K 10 ----
  for (int mt = 0; mt < 2; ++mt) {
    for (int nt = 0; nt < 2; ++nt) {
      v8f acc = {};
      const int col = nt * 16 + ncol;
      for (int k = 0; k < 10; ++k) {
        const int row = mt * 16 + mrow;
        const int kr  = k * 4 + half * 2;
        v2f a, b;
        a.x = h[row][kr + 0];
        a.y = h[row][kr + 1];
        b.x = (col < H2_) ? w2[(kr + 0) * H2_ + col] : 0.0f;
        b.y = (col < H2_) ? w2[(kr + 1) * H2_ + col] : 0.0f;
        acc = __builtin_amdgcn_wmma_f32_16x16x4_f32(
            false, a, false, b, (short)0, acc, false, false);
      }
      if (col < H2_) {
        const float bias = b2[col];
        for (int v = 0; v < 8; ++v) {
          const int row = mt * 16 + v + half * 8;
          dout[row * H2_ + col] = fmaxf(acc[v] + bias, 0.0f);
        }
      }
    }
  }
}

// ---------------------------------------------------------------------------
// Kernel 2 (pass A): per-chunk aggregates (max, argmax, sum) for each (n,c).
// Grid (NCH_, N_), 256 threads (= C). Strict '>' so first max wins (ref ties).
// ---------------------------------------------------------------------------
__global__ __launch_bounds__(256) void chunk_agg_kernel(
    const float* __restrict__ x, float* __restrict__ aggMax,
    float* __restrict__ aggSum, int* __restrict__ aggIdx) {
  const int k = blockIdx.x, n = blockIdx.y, c = threadIdx.x;
  const float* xp = x + ((size_t)n * L_) * C_ + c;
  float m = -INFINITY, s = 0.0f;
  int mi = 0;
  const int t0 = k * CL_;
#pragma unroll 4
  for (int tt = 0; tt < CL_; ++tt) {
    const int t = t0 + tt;
    const float xt = xp[(size_t)t * C_];
    if (xt > m) { m = xt; mi = t; }
    s += xt;
  }
  const size_t ai = ((size_t)n * NCH_ + k) * C_ + c;
  aggMax[ai] = m;
  aggSum[ai] = s;
  aggIdx[ai] = mi;
}

// ---------------------------------------------------------------------------
// Kernel 3 (pass B): combine chunk prefixes, rescan own chunk, and fuse the
// three LayerNorms per timestep (block == full C dim), writing final outputs.
// Cross-lane stats: wave32 shfl_xor butterfly + one cross-wave LDS combine,
// double-buffered so only ONE barrier per timestep is needed.
// ---------------------------------------------------------------------------
__global__ __launch_bounds__(256) void scan_ln_kernel(
    const float* __restrict__ x, const float* __restrict__ aggMax,
    const float* __restrict__ aggSum, const int* __restrict__ aggIdx,
    const float* __restrict__ gam, const float* __restrict__ bet,
    const float* __restrict__ dmlp, float* __restrict__ out,
    float* __restrict__ idx_out, float* __restrict__ pmax_out) {
  __shared__ float part[2][8][6];   // [buffer][wave][sum/sumsq x3]

  const int k = blockIdx.x, n = blockIdx.y, c = threadIdx.x;
  const int wave = c >> 5, lane = c & 31;

  // Prefix over chunks 0..k-1 (tiny: <= 15 iterations).
  float m = -INFINITY, s = 0.0f;
  int mi = 0;
  for (int j = 0; j < k; ++j) {
    const size_t ai = ((size_t)n * NCH_ + j) * C_ + c;
    const float cm = aggMax[ai];
    if (cm > m) { m = cm; mi = aggIdx[ai]; }
    s += aggSum[ai];
  }

  const float g0 = gam[c],        b0 = bet[c];
  const float g1 = gam[C_ + c],   b1 = bet[C_ + c];
  const float g2 = gam[2*C_ + c], b2 = bet[2*C_ + c];
  const float dv = (c < H2_) ? dmlp[n * H2_ + c] : 0.0f;

  const float* xp = x + ((size_t)n * L_) * C_ + c;
  const int t0 = k * CL_;

  for (int tt = 0; tt < CL_; ++tt) {
    const int t = t0 + tt;
    const float xt = xp[(size_t)t * C_];
    if (xt > m) { m = xt; mi = t; }
    s += xt;

    // Pool values (pad-zero wins the max for t < L-1 when running max < 0).
    const bool pw = (t < L_ - 1) && (m < 0.0f);
    const float pm = pw ? 0.0f : m;
    const int   pi = pw ? -1 : mi;
    const float pa = s * (1.0f / (float)L_);
    const float pq = s * 0.022097086912079608f;  // 1/sqrt(2048)

    // Wave-level butterfly reductions of sum / sum-of-squares (x3 tensors).
    float r0 = pm, r1 = pm * pm, r2 = pa, r3 = pa * pa, r4 = pq, r5 = pq * pq;
#pragma unroll
    for (int off = 16; off > 0; off >>= 1) {
      r0 += __shfl_xor(r0, off);
      r1 += __shfl_xor(r1, off);
      r2 += __shfl_xor(r2, off);
      r3 += __shfl_xor(r3, off);
      r4 += __shfl_xor(r4, off);
      r5 += __shfl_xor(r5, off);
    }
    const int p = tt & 1;
    if (lane == 0) {
      part[p][wave][0] = r0; part[p][wave][1] = r1; part[p][wave][2] = r2;
      part[p][wave][3] = r3; part[p][wave][4] = r4; part[p][wave][5] = r5;
    }
    __syncthreads();
    float s0 = 0, q0 = 0, s1 = 0, q1 = 0, s2 = 0, q2 = 0;
#pragma unroll
    for (int w = 0; w < 8; ++w) {
      s0 += part[p][w][0]; q0 += part[p][w][1];
      s1 += part[p][w][2]; q1 += part[p][w][3];
      s2 += part[p][w][4]; q2 += part[p][w][5];
    }
    const float inv = 1.0f / (float)C_;
    const float mu0 = s0 * inv, v0 = fmaxf(q0 * inv - mu0 * mu0, 0.0f);
    const float mu1 = s1 * inv, v1 = fmaxf(q1 * inv - mu1 * mu1, 0.0f);
    const float mu2 = s2 * inv, v2 = fmaxf(q2 * inv - mu2 * mu2, 0.0f);

    const size_t row = (size_t)n * L_ + t;
    const size_t ob  = row * OC_;
    out[ob + c]          = (pm - mu0) * rsqrtf(v0 + 1e-5f) * g0 + b0;
    out[ob + C_ + c]     = (pa - mu1) * rsqrtf(v1 + 1e-5f) * g1 + b1;
    out[ob + 2*C_ + c]   = (pq - mu2) * rsqrtf(v2 + 1e-5f) * g2 + b2;
    if (c < H2_) out[ob + 3*C_ + c] = dv;
    idx_out[row * C_ + c]  = (float)pi;
    pmax_out[row * C_ + c] = pm;
  }
}

// ---------------------------------------------------------------------------
// Launch: MLP (WMMA) -> chunk aggregates -> fused scan + LayerNorm + concat.
// ---------------------------------------------------------------------------
extern "C" void kernel_launch(void* const* d_in, const int* in_sizes, int n_in,
                              void* d_out, int out_size, void* d_ws, size_t ws_size,
                              hipStream_t stream) {
  const float* dem = (const float*)d_in[0];
  const float* x   = (const float*)d_in[1];
  const float* gam = (const float*)d_in[2];
  const float* bet = (const float*)d_in[3];
  const float* w1  = (const float*)d_in[4];
  const float* b1  = (const float*)d_in[5];
  const float* w2  = (const float*)d_in[6];
  const float* b2  = (const float*)d_in[7];

  float* out      = (float*)d_out;                       // (N,L,788)
  float* idx_out  = out + (size_t)N_ * L_ * OC_;         // (N,L,256) indices
  float* pmax_out = idx_out + (size_t)N_ * L_ * C_;      // (N,L,256) raw p_max

  // Workspace layout (all 4-byte words): [ d(32x20) pad->1024 | aggMax | aggSum | aggIdx ]
  float* dmlp   = (float*)d_ws;
  float* aggMax = dmlp + 1024;
  float* aggSum = aggMax + (size_t)N_ * NCH_ * C_;
  int*   aggIdx = (int*)(aggSum + (size_t)N_ * NCH_ * C_);

  mlp_wmma_kernel<<<1, 32, 0, stream>>>(dem, w1, b1, w2, b2, dmlp);
  chunk_agg_kernel<<<dim3(NCH_, N_), C_, 0, stream>>>(x, aggMax, aggSum, aggIdx);
  scan_ln_kernel<<<dim3(NCH_, N_), C_, 0, stream>>>(
      x, aggMax, aggSum, aggIdx, gam, bet, dmlp, out, idx_out, pmax_out);
}